// ClusteringLayer_76871324664002
// MI455X (gfx1250) — compile-verified
//
#include <hip/hip_runtime.h>
#include <hip/hip_bf16.h>

typedef float v2f __attribute__((ext_vector_type(2)));
typedef float v8f __attribute__((ext_vector_type(8)));

#define D_DIM      512
#define K_CENT     4096
#define N_ROWS     (8 * 2048)
#define M_TILE     64          // rows of x per block
#define NCHUNK     32          // centroid columns per chunk
#define LDS_STRIDE 516         // padded row stride (floats): 516 % 64 = 4 -> conflict-free frags
#define THREADS    256
#define BIG_F      3.402823466e38f

// ---------------------------------------------------------------------------
// CDNA5 async copy: global -> LDS, 16B per lane, tracked by ASYNCcnt.
// ldsoff = low 32 bits of the generic LDS pointer.
// ---------------------------------------------------------------------------
__device__ __forceinline__ void async_load_b128(const void* gaddr, unsigned int ldsoff)
{
    asm volatile("global_load_async_to_lds_b128 %0, %1, off"
                 :: "v"(ldsoff), "v"(gaddr)
                 : "memory");
}

__device__ __forceinline__ void wait_async_all()
{
    asm volatile("s_wait_asynccnt 0" ::: "memory");
}

// stage one 32x512 centroid chunk (4096 float4) with async copies
__device__ __forceinline__ void stage_chunk_async(const float* __restrict__ cent,
                                                  int kb, float* csbuf, int t)
{
    #pragma unroll
    for (int i = 0; i < (NCHUNK * D_DIM / 4) / THREADS; ++i) {   // 16 iters
        int idx = i * THREADS + t;
        int row = idx >> 7;            // 128 float4 per centroid row
        int dv  = idx & 127;
        const void* g = (const void*)(cent + (size_t)(kb + row) * D_DIM + dv * 4);
        unsigned int l = (unsigned int)(uintptr_t)(csbuf + row * LDS_STRIDE + dv * 4);
        async_load_b128(g, l);
    }
}

// ---------------------------------------------------------------------------
// Nearest-centroid assignment: 64-row x tile resident in LDS, centroids
// streamed in double-buffered async chunks of 32, fp32 WMMA dot products,
// distributed running argmin.
// ---------------------------------------------------------------------------
__global__ __launch_bounds__(THREADS)
void kmeans_assign_kernel(const float* __restrict__ x,
                          const float* __restrict__ cent,
                          float* __restrict__ out_y)
{
    extern __shared__ float smem[];
    float* xs   = smem;                               // [M_TILE][LDS_STRIDE]
    float* cs0  = xs  + M_TILE * LDS_STRIDE;          // [NCHUNK][LDS_STRIDE]
    float* cs1  = cs0 + NCHUNK * LDS_STRIDE;          // [NCHUNK][LDS_STRIDE]
    float* csq  = cs1 + NCHUNK * LDS_STRIDE;          // [NCHUNK]
    float* psum = csq + NCHUNK;                       // [THREADS]
    float* redV = psum + THREADS;                     // [M_TILE][NCHUNK]
    int*   redI = (int*)(redV + M_TILE * NCHUNK);     // [M_TILE][NCHUNK]

    const int t    = threadIdx.x;
    const int lane = t & 31;           // wave32
    const int wave = t >> 5;           // 8 waves
    const int hi   = lane >> 4;        // 0: K=0,1  1: K=2,3 (A/B frag halves)
    const int l16  = lane & 15;
    const int mrow = (wave >> 1) * 16; // wave row offset in tile   (4 groups)
    const int ncol = (wave & 1) * 16;  // wave col offset in chunk  (2 groups)

    const int blockRow = blockIdx.x * M_TILE;

    // ---- prologue: async-stage x tile (64x512) and centroid chunk 0 ----
    #pragma unroll
    for (int i = 0; i < (M_TILE * D_DIM / 4) / THREADS; ++i) {   // 32 iters
        int idx = i * THREADS + t;
        int row = idx >> 7;            // 128 float4 per row
        int dv  = idx & 127;
        const void* g = (const void*)(x + (size_t)(blockRow + row) * D_DIM + dv * 4);
        unsigned int l = (unsigned int)(uintptr_t)(xs + row * LDS_STRIDE + dv * 4);
        async_load_b128(g, l);
    }
    stage_chunk_async(cent, 0, cs0, t);
    wait_async_all();
    __syncthreads();

    // per-lane fragment base pointers (layouts per CDNA5 ISA 7.12.2)
    // A 16x4 f32: lane l16 -> M=l16, vgpr pair holds K = hi*2, hi*2+1
    // B 4x16 f32: lane l16 -> N=l16, vgpr pair holds K = hi*2, hi*2+1
    const float* aPtr  = xs  + (mrow + l16) * LDS_STRIDE + hi * 2;
    const float* bPtr0 = cs0 + (ncol + l16) * LDS_STRIDE + hi * 2;
    const float* bPtr1 = cs1 + (ncol + l16) * LDS_STRIDE + hi * 2;

    float minv[8];
    int   mini[8];
    #pragma unroll
    for (int r = 0; r < 8; ++r) { minv[r] = BIG_F; mini[r] = 0; }

    for (int kb = 0; kb < K_CENT; kb += NCHUNK) {
        const int buf = (kb / NCHUNK) & 1;
        float* cs = buf ? cs1 : cs0;
        const float* bPtr = buf ? bPtr1 : bPtr0;

        // kick off the next chunk into the other buffer (overlaps compute)
        if (kb + NCHUNK < K_CENT)
            stage_chunk_async(cent, kb + NCHUNK, buf ? cs0 : cs1, t);

        // ---- sum of centroid elements for the current chunk, from LDS ----
        {
            int n   = t >> 3;   // 0..31
            int seg = t & 7;    // 8 segments of 64
            const float* p = cs + n * LDS_STRIDE + seg * 64;
            float s = 0.f;
            #pragma unroll
            for (int q = 0; q < 64; ++q) s += p[q];
            psum[t] = s;
        }
        __syncthreads();
        if (t < NCHUNK) {
            float s = 0.f;
            #pragma unroll
            for (int q = 0; q < 8; ++q) s += psum[t * 8 + q];
            csq[t] = s;
        }
        __syncthreads();

        // ---- fp32 WMMA: 16x16 dot tile over D=512 (128 x K=4 steps) ----
        v8f acc = {0.f, 0.f, 0.f, 0.f, 0.f, 0.f, 0.f, 0.f};
        #pragma unroll 8
        for (int d = 0; d < D_DIM; d += 4) {
            v2f a = *(const v2f*)(aPtr + d);
            v2f b = *(const v2f*)(bPtr + d);
            acc = __builtin_amdgcn_wmma_f32_16x16x4_f32(
                /*neg_a=*/false, a, /*neg_b=*/false, b,
                /*c_mod=*/(short)0, acc, /*reuse_a=*/false, /*reuse_b=*/false);
        }

        // ---- dist = sum(c) - 2*dot ; running argmin per lane-slot ----
        float cq   = csq[ncol + l16];
        int   cidx = kb + ncol + l16;
        #pragma unroll
        for (int r = 0; r < 8; ++r) {
            float dist = __builtin_fmaf(-2.0f, acc[r], cq);
            if (dist < minv[r]) { minv[r] = dist; mini[r] = cidx; }
        }

        // close pipeline stage: our async loads done, everyone done reading cs
        wait_async_all();
        __syncthreads();
    }

    // ---- cross-slot reduction: each row has 32 candidates (2 waves x 16) ----
    #pragma unroll
    for (int r = 0; r < 8; ++r) {
        int m    = mrow + hi * 8 + r;   // C/D layout: vgpr r -> M = r or 8+r
        int slot = ncol + l16;
        redV[m * NCHUNK + slot] = minv[r];
        redI[m * NCHUNK + slot] = mini[r];
    }
    __syncthreads();

    if (t < M_TILE) {
        float bv = BIG_F;
        int   bi = 0x7fffffff;
        #pragma unroll 4
        for (int j = 0; j < NCHUNK; ++j) {
            float v   = redV[t * NCHUNK + j];
            int   idx = redI[t * NCHUNK + j];
            if (v < bv || (v == bv && idx < bi)) { bv = v; bi = idx; }
        }
        out_y[blockRow + t] = (float)bi;   // first-occurrence argmin semantics
    }
}

// ---------------------------------------------------------------------------
// Pass-through copy of x into the first N*D slots of the tuple output.
// ---------------------------------------------------------------------------
__global__ __launch_bounds__(256)
void copy_x_kernel(const float4* __restrict__ src, float4* __restrict__ dst, int n4)
{
    int i      = blockIdx.x * blockDim.x + threadIdx.x;
    int stride = gridDim.x * blockDim.x;
    for (; i < n4; i += stride) dst[i] = src[i];
}

extern "C" void kernel_launch(void* const* d_in, const int* in_sizes, int n_in,
                              void* d_out, int out_size, void* d_ws, size_t ws_size,
                              hipStream_t stream)
{
    const float* x    = (const float*)d_in[0];   // [8, 2048, 512] f32
    const float* cent = (const float*)d_in[1];   // [4096, 512]    f32
    float* out = (float*)d_out;                  // x flat (8388608) ++ y (16384)

    const size_t xElems = (size_t)N_ROWS * D_DIM;

    // output[0]: x copied verbatim
    copy_x_kernel<<<1024, 256, 0, stream>>>(
        (const float4*)x, (float4*)out, (int)(xElems / 4));

    // output[1]: nearest-centroid indices
    size_t lds = (size_t)(M_TILE + 2 * NCHUNK) * LDS_STRIDE * sizeof(float)   // xs + cs0 + cs1
               + NCHUNK * sizeof(float)                                       // csq
               + THREADS * sizeof(float)                                      // psum
               + (size_t)M_TILE * NCHUNK * (sizeof(float) + sizeof(int));     // reduction
    kmeans_assign_kernel<<<N_ROWS / M_TILE, THREADS, lds, stream>>>(
        x, cent, out + xElems);
}